// TypeDP_83176336655080
// MI455X (gfx1250) — compile-verified
//
#include <hip/hip_runtime.h>
#include <math.h>

// ---------------- problem constants (from reference) ----------------
constexpr int N_    = 4096;
constexpr int NT_   = 2;
constexpr int MAXN_ = 100;
constexpr int M2_   = 16;
constexpr int TF_   = 4;
constexpr int NA_   = N_ / NT_;     // 2048
constexpr int J_    = NT_ * MAXN_;  // 200
constexpr int GD_   = 100;          // GDIM
constexpr int D_    = M2_ * GD_;    // 1600
constexpr int FH_   = 240;

typedef __attribute__((ext_vector_type(2))) float v2f;
typedef __attribute__((ext_vector_type(8))) float v8f;

// =====================================================================
// Tensor Data Mover helper: 1-D f32 copy global -> LDS (ISA ch. 8 D#).
// Group0: count=1 | lds_addr | global_addr[56:0] | type=2.
// Group1: data_size=4B, tensor_dim0 = tile_dim0 = nelem (<=65535),
//         tensor_dim1 = tile_dim1 = 1, stride0 = nelem.
// Tracked by TENSORcnt -> s_wait_tensorcnt 0.
// =====================================================================
#if __has_builtin(__builtin_amdgcn_tensor_load_to_lds)
#define HAVE_TDM 1
__device__ __forceinline__ void tdm_load_f32_1d(unsigned lds_byte_off,
                                                const float* gptr,
                                                unsigned nelem)
{
  typedef unsigned int u32x4 __attribute__((ext_vector_type(4)));
  typedef int i32x8 __attribute__((ext_vector_type(8)));
  typedef int i32x4 __attribute__((ext_vector_type(4)));
  const unsigned long long ga = (unsigned long long)(uintptr_t)gptr;
  u32x4 g0;
  g0.x = 1u;                                               // count=1 (valid D#)
  g0.y = lds_byte_off;                                     // lds_addr
  g0.z = (unsigned)(ga & 0xFFFFFFFFu);                     // global_addr[31:0]
  g0.w = (unsigned)((ga >> 32) & 0x01FFFFFFu) | (2u << 30);// addr[56:32] | type=2
  i32x8 g1;
  g1[0] = (2 << 16);                                       // data_size = 4 bytes
  g1[1] = (int)((nelem & 0xFFFFu) << 16);                  // tensor_dim0[15:0]
  g1[2] = (int)(((nelem >> 16) & 0xFFFFu) | (1u << 16));   // dim0[31:16] | tensor_dim1=1
  g1[3] = (int)((nelem & 0xFFFFu) << 16);                  // tile_dim0
  g1[4] = 1;                                               // tile_dim1 = 1
  g1[5] = (int)nelem;                                      // tensor_dim0_stride lo32
  g1[6] = 0;
  g1[7] = 0;
  i32x4 z4 = {0,0,0,0};
#if defined(__clang_major__) && __clang_major__ >= 23
  i32x8 z8 = {0,0,0,0,0,0,0,0};
  __builtin_amdgcn_tensor_load_to_lds(g0, g1, z4, z4, z8, 0);
#else
  __builtin_amdgcn_tensor_load_to_lds(g0, g1, z4, z4, 0);
#endif
}
#else
#define HAVE_TDM 0
#endif

// =====================================================================
// Kernel 1: embedding forward + xa contraction + Gram features (per atom)
// One 256-thread block per atom. Weights (26KB) + G (80KB) staged in LDS
// (CDNA5 WGP has 320KB LDS).
// =====================================================================
__global__ __launch_bounds__(256) void emb_fwd(
    const float* __restrict__ Ri, const float* __restrict__ tv,
    const float* __restrict__ ew0, const float* __restrict__ eb0,
    const float* __restrict__ ew1, const float* __restrict__ eb1,
    const float* __restrict__ ew2, const float* __restrict__ eb2,
    float* __restrict__ xa_ws, float* __restrict__ dr_ws)
{
  __shared__ float sW0[5*25];
  __shared__ float sB0[25];
  __shared__ float sW1[25*50];
  __shared__ float sB1[50];
  __shared__ float sW2[50*GD_];
  __shared__ float sB2[GD_];
  __shared__ float sA[J_*4];
  __shared__ float sG[J_*GD_];     // 80 KB
  __shared__ float sXA[4*GD_];

  const int n   = blockIdx.x;
  const int tid = threadIdx.x;

  for (int i=tid;i<5*25;i+=256)   sW0[i]=ew0[i];
  for (int i=tid;i<25;i+=256)     sB0[i]=eb0[i];
  for (int i=tid;i<25*50;i+=256)  sW1[i]=ew1[i];
  for (int i=tid;i<50;i+=256)     sB1[i]=eb1[i];
  for (int i=tid;i<50*GD_;i+=256) sW2[i]=ew2[i];
  for (int i=tid;i<GD_;i+=256)    sB2[i]=eb2[i];
  for (int i=tid;i<J_*4;i+=256)   sA[i]=Ri[(size_t)n*J_*4 + i];
  __syncthreads();

  if (tid < J_) {
    const int j  = tid;
    const int tt = j / MAXN_;
    float x[5];
    x[0] = sA[j*4+0];
    #pragma unroll
    for (int f=0;f<TF_;f++) x[1+f] = tv[tt*TF_+f];
    float h0[25];
    #pragma unroll
    for (int i=0;i<25;i++){
      float acc = sB0[i];
      #pragma unroll
      for (int c=0;c<5;c++) acc += x[c]*sW0[c*25+i];
      h0[i]=tanhf(acc);
    }
    float h1[50];
    #pragma unroll
    for (int k=0;k<50;k++){
      float acc=sB1[k];
      #pragma unroll
      for (int i=0;i<25;i++) acc += h0[i]*sW1[i*50+k];
      h1[k]=tanhf(acc);
    }
    for (int g=0; g<GD_; g++){
      float acc=sB2[g];
      #pragma unroll
      for (int k=0;k<50;k++) acc += h1[k]*sW2[k*GD_+g];
      sG[j*GD_+g]=tanhf(acc);
    }
  }
  __syncthreads();

  const float inv = 1.0f/(float)(MAXN_*NT_);
  for (int idx=tid; idx<4*GD_; idx+=256){
    const int c = idx / GD_;
    const int g = idx % GD_;
    float acc=0.f;
    for (int j=0;j<J_;j++) acc += sA[j*4+c]*sG[j*GD_+g];
    acc *= inv;
    sXA[idx]=acc;
    xa_ws[(size_t)n*4*GD_ + idx]=acc;
  }
  __syncthreads();

  for (int idx=tid; idx<D_; idx+=256){
    const int g = idx >> 4;
    const int m = idx & 15;
    float acc=0.f;
    #pragma unroll
    for (int c=0;c<4;c++) acc += sXA[c*GD_+g]*sXA[c*GD_+m];
    dr_ws[(size_t)n*D_ + idx]=acc;
  }
}

// =====================================================================
// WMMA f32 16x16x4 GEMM kernels.
// A lane layout (ISA 7.12.2): lanes 0-15 row=lane K={0,1}, lanes 16-31
// row=lane-16 K={2,3}; B mirrored; C: VGPR i -> M=i / M=i+8, N=lane&15.
// =====================================================================

// H1 = tanh(DR @ fw0[t] + fb0[t]); M=4096, K=1600, N=240.
// One wave per 16-row x 80-col strip (5 N-tiles) to amortize A loads 5x.
__global__ __launch_bounds__(32) void fit_l0(
    const float* __restrict__ A_ws, const float* __restrict__ fw0,
    const float* __restrict__ fb0, float* __restrict__ h1_ws)
{
  const int mt = blockIdx.x;            // 0..255 (16-row tiles)
  const int ns = blockIdx.y;            // 0..2   (strips of 5 N-tiles)
  const int lane = threadIdx.x;
  const int t    = (mt*16) / NA_;
  const int arow = lane & 15;
  const int kh   = (lane >> 4) * 2;
  const float* __restrict__ A  = A_ws + (size_t)mt*16*D_;
  const float* __restrict__ Bw = fw0 + (size_t)t*D_*FH_ + (size_t)ns*80;
  __builtin_prefetch(Bw, 0, 3);
  v8f c[5];
  const v8f z = {};
  #pragma unroll
  for (int s=0;s<5;s++) c[s]=z;
  for (int k=0;k<D_;k+=4){
    v2f a;
    a.x = A[(size_t)arow*D_ + (k+kh)];
    a.y = A[(size_t)arow*D_ + (k+kh+1)];
    #pragma unroll
    for (int s=0;s<5;s++){
      v2f b;
      b.x = Bw[(size_t)(k+kh)*FH_ + s*16 + arow];
      b.y = Bw[(size_t)(k+kh+1)*FH_ + s*16 + arow];
      c[s] = __builtin_amdgcn_wmma_f32_16x16x4_f32(false, a, false, b, (short)0, c[s], false, false);
    }
  }
  const int rbase = mt*16 + ((lane>>4)*8);
  #pragma unroll
  for (int s=0;s<5;s++){
    const int col = ns*80 + s*16 + (lane & 15);
    const float bias = fb0[t*FH_ + col];
    #pragma unroll
    for (int i=0;i<8;i++)
      h1_ws[(size_t)(rbase+i)*FH_ + col] = tanhf(c[s][i]+bias);
  }
}

// H2 = tanh(H1 @ fw1[t] + fb1[t]); K=240, N=240.
// 8-wave block (128 rows). Whole fw1[t] panel (230KB) staged into the
// 320KB WGP LDS by the Tensor Data Mover; each wave owns a full 16x240
// row strip (15 N-tiles, 120 accumulator VGPRs).
__global__ __launch_bounds__(256) void fit_l1(
    const float* __restrict__ h1_ws, const float* __restrict__ fw1,
    const float* __restrict__ fb1, float* __restrict__ h2_ws)
{
  __shared__ float sB[FH_*FH_];   // 230,400 B; single shared array -> LDS offset 0
  const int t = (blockIdx.x*128) / NA_;
  const float* __restrict__ Bsrc = fw1 + (size_t)t*FH_*FH_;
#if HAVE_TDM
  if (threadIdx.x == 0) {
    tdm_load_f32_1d(0u, Bsrc, FH_*FH_);
    __builtin_amdgcn_s_wait_tensorcnt(0);
  }
#else
  for (int i = threadIdx.x; i < FH_*FH_; i += 256) sB[i] = Bsrc[i];
#endif
  __syncthreads();

  const int wave = threadIdx.x >> 5;
  const int lane = threadIdx.x & 31;
  const int mt   = blockIdx.x*8 + wave;
  const int arow = lane & 15;
  const int kh   = (lane >> 4) * 2;
  const float* __restrict__ A = h1_ws + (size_t)mt*16*FH_;

  v8f c[15];
  const v8f z = {};
  #pragma unroll
  for (int nt=0;nt<15;nt++) c[nt]=z;
  for (int k=0;k<FH_;k+=4){
    v2f a;
    a.x = A[(size_t)arow*FH_ + (k+kh)];
    a.y = A[(size_t)arow*FH_ + (k+kh+1)];
    #pragma unroll
    for (int nt=0;nt<15;nt++){
      v2f b;
      b.x = sB[(k+kh)*FH_ + nt*16 + arow];
      b.y = sB[(k+kh+1)*FH_ + nt*16 + arow];
      c[nt] = __builtin_amdgcn_wmma_f32_16x16x4_f32(false, a, false, b, (short)0, c[nt], false, false);
    }
  }
  const int rbase = mt*16 + ((lane>>4)*8);
  #pragma unroll
  for (int nt=0;nt<15;nt++){
    const int col = nt*16 + (lane & 15);
    const float bias = fb1[t*FH_ + col];
    #pragma unroll
    for (int i=0;i<8;i++)
      h2_ws[(size_t)(rbase+i)*FH_ + col] = tanhf(c[nt][i]+bias);
  }
}

// Ei = H2 . fw2[t] + fb2[t]; also Etot via atomic
__global__ __launch_bounds__(32) void fit_out_k(
    const float* __restrict__ h2_ws, const float* __restrict__ fw2,
    const float* __restrict__ fb2, float* __restrict__ out)
{
  const int n = blockIdx.x, lane = threadIdx.x;
  const int t = n / NA_;
  float acc = 0.f;
  for (int k=lane;k<FH_;k+=32) acc += h2_ws[(size_t)n*FH_+k]*fw2[t*FH_+k];
  #pragma unroll
  for (int off=16;off>0;off>>=1) acc += __shfl_down(acc, off, 32);
  if (lane==0){
    float e = acc + fb2[t];
    out[1+n] = e;
    atomicAdd(&out[0], e);
  }
}

// dz2 = fw2[t] * (1 - h2^2)
__global__ __launch_bounds__(256) void dz2_k(
    const float* __restrict__ h2_ws, const float* __restrict__ fw2,
    float* __restrict__ dz2_ws)
{
  const int i = blockIdx.x*256 + threadIdx.x;
  if (i >= N_*FH_) return;
  const int n = i / FH_, k = i % FH_;
  const int t = n / NA_;
  const float h = h2_ws[i];
  dz2_ws[i] = fw2[t*FH_+k]*(1.f-h*h);
}

// dz1 = (dz2 @ fw1[t]^T) * (1 - h1^2); K=240, N=240.
// Same TDM-staged LDS panel, B read transposed from LDS.
__global__ __launch_bounds__(256) void bwd_l1(
    const float* __restrict__ dz2_ws, const float* __restrict__ fw1,
    const float* __restrict__ h1_ws, float* __restrict__ dz1_ws)
{
  __shared__ float sB[FH_*FH_];
  const int t = (blockIdx.x*128) / NA_;
  const float* __restrict__ Bsrc = fw1 + (size_t)t*FH_*FH_;
#if HAVE_TDM
  if (threadIdx.x == 0) {
    tdm_load_f32_1d(0u, Bsrc, FH_*FH_);
    __builtin_amdgcn_s_wait_tensorcnt(0);
  }
#else
  for (int i = threadIdx.x; i < FH_*FH_; i += 256) sB[i] = Bsrc[i];
#endif
  __syncthreads();

  const int wave = threadIdx.x >> 5;
  const int lane = threadIdx.x & 31;
  const int mt   = blockIdx.x*8 + wave;
  const int arow = lane & 15;
  const int kh   = (lane >> 4) * 2;
  const float* __restrict__ A = dz2_ws + (size_t)mt*16*FH_;

  v8f c[15];
  const v8f z = {};
  #pragma unroll
  for (int nt=0;nt<15;nt++) c[nt]=z;
  for (int k=0;k<FH_;k+=4){
    v2f a;
    a.x = A[(size_t)arow*FH_ + (k+kh)];
    a.y = A[(size_t)arow*FH_ + (k+kh+1)];
    #pragma unroll
    for (int nt=0;nt<15;nt++){
      v2f b;
      b.x = sB[(nt*16 + arow)*FH_ + (k+kh)];     // fw1^T
      b.y = sB[(nt*16 + arow)*FH_ + (k+kh+1)];
      c[nt] = __builtin_amdgcn_wmma_f32_16x16x4_f32(false, a, false, b, (short)0, c[nt], false, false);
    }
  }
  const int rbase = mt*16 + ((lane>>4)*8);
  #pragma unroll
  for (int nt=0;nt<15;nt++){
    const int col = nt*16 + (lane & 15);
    #pragma unroll
    for (int i=0;i<8;i++){
      const size_t o = (size_t)(rbase+i)*FH_ + col;
      const float h = h1_ws[o];
      dz1_ws[o] = c[nt][i]*(1.f-h*h);
    }
  }
}

// d_dr = dz1 @ fw0[t]^T; K=240, N=1600 (B transposed), 5-tile N strips.
__global__ __launch_bounds__(32) void bwd_l0(
    const float* __restrict__ dz1_ws, const float* __restrict__ fw0,
    float* __restrict__ ddr_ws)
{
  const int mt = blockIdx.x;            // 0..255
  const int ns = blockIdx.y;            // 0..19 (strips of 5 N-tiles)
  const int lane = threadIdx.x;
  const int t    = (mt*16) / NA_;
  const int arow = lane & 15;
  const int kh   = (lane >> 4) * 2;
  const float* __restrict__ A  = dz1_ws + (size_t)mt*16*FH_;
  const float* __restrict__ Bw = fw0 + (size_t)t*D_*FH_;
  __builtin_prefetch(Bw + (size_t)ns*80*FH_, 0, 3);
  v8f c[5];
  const v8f z = {};
  #pragma unroll
  for (int s=0;s<5;s++) c[s]=z;
  for (int k=0;k<FH_;k+=4){
    v2f a;
    a.x = A[(size_t)arow*FH_ + (k+kh)];
    a.y = A[(size_t)arow*FH_ + (k+kh+1)];
    #pragma unroll
    for (int s=0;s<5;s++){
      const int bcol = ns*80 + s*16 + arow;
      v2f b;
      b.x = Bw[(size_t)bcol*FH_ + (k+kh)];       // fw0^T
      b.y = Bw[(size_t)bcol*FH_ + (k+kh+1)];
      c[s] = __builtin_amdgcn_wmma_f32_16x16x4_f32(false, a, false, b, (short)0, c[s], false, false);
    }
  }
  const int rbase = mt*16 + ((lane>>4)*8);
  #pragma unroll
  for (int s=0;s<5;s++){
    const int col = ns*80 + s*16 + (lane & 15);
    #pragma unroll
    for (int i=0;i<8;i++)
      ddr_ws[(size_t)(rbase+i)*D_ + col] = c[s][i];
  }
}

// =====================================================================
// Embedding backward (recompute activations) fused with dE_dx, force
// self-term + scatter-add, and virial reduction.
// =====================================================================
__global__ __launch_bounds__(256) void emb_bwd(
    const float* __restrict__ Ri, const float* __restrict__ tv,
    const float* __restrict__ ew0, const float* __restrict__ eb0,
    const float* __restrict__ ew1, const float* __restrict__ eb1,
    const float* __restrict__ ew2, const float* __restrict__ eb2,
    const float* __restrict__ dfeat, const float* __restrict__ ImDR,
    const int*   __restrict__ list_neigh,
    const float* __restrict__ xa_ws, const float* __restrict__ ddr_ws,
    float* __restrict__ out)
{
  __shared__ float sW0[5*25];
  __shared__ float sB0[25];
  __shared__ float sW1[25*50];
  __shared__ float sB1[50];
  __shared__ float sW2[50*GD_];
  __shared__ float sB2[GD_];
  __shared__ float sA[J_*4];
  __shared__ float sXA[4*GD_];
  __shared__ float sDXA[4*GD_];
  __shared__ float sDDR[D_];
  __shared__ float sF[3];
  __shared__ float sV[6];

  const int n   = blockIdx.x;
  const int tid = threadIdx.x;

  for (int i=tid;i<5*25;i+=256)   sW0[i]=ew0[i];
  for (int i=tid;i<25;i+=256)     sB0[i]=eb0[i];
  for (int i=tid;i<25*50;i+=256)  sW1[i]=ew1[i];
  for (int i=tid;i<50;i+=256)     sB1[i]=eb1[i];
  for (int i=tid;i<50*GD_;i+=256) sW2[i]=ew2[i];
  for (int i=tid;i<GD_;i+=256)    sB2[i]=eb2[i];
  for (int i=tid;i<J_*4;i+=256)   sA[i]=Ri[(size_t)n*J_*4 + i];
  for (int i=tid;i<4*GD_;i+=256)  sXA[i]=xa_ws[(size_t)n*4*GD_ + i];
  for (int i=tid;i<D_;i+=256)     sDDR[i]=ddr_ws[(size_t)n*D_ + i];
  if (tid<3) sF[tid]=0.f;
  if (tid<6) sV[tid]=0.f;
  __syncthreads();

  // d_xa[c,q] = sum_m ddr[q,m]*xa[c,m]  +  (q<16) sum_g ddr[g,q]*xa[c,g]
  for (int idx=tid; idx<4*GD_; idx+=256){
    const int c = idx / GD_;
    const int q = idx % GD_;
    float acc = 0.f;
    #pragma unroll
    for (int m=0;m<M2_;m++) acc += sDDR[q*M2_+m]*sXA[c*GD_+m];
    if (q < M2_){
      for (int g=0;g<GD_;g++) acc += sDDR[g*M2_+q]*sXA[c*GD_+g];
    }
    sDXA[idx]=acc;
  }
  __syncthreads();

  if (tid < J_){
    const int j  = tid;
    const int tt = j / MAXN_;
    const float inv = 1.0f/(float)(MAXN_*NT_);
    float x[5];
    x[0] = sA[j*4+0];
    #pragma unroll
    for (int f=0;f<TF_;f++) x[1+f] = tv[tt*TF_+f];
    float h0[25];
    #pragma unroll
    for (int i=0;i<25;i++){
      float acc = sB0[i];
      #pragma unroll
      for (int c=0;c<5;c++) acc += x[c]*sW0[c*25+i];
      h0[i]=tanhf(acc);
    }
    float h1e[50];
    #pragma unroll
    for (int k=0;k<50;k++){
      float acc=sB1[k];
      #pragma unroll
      for (int i=0;i<25;i++) acc += h0[i]*sW1[i*50+k];
      h1e[k]=tanhf(acc);
    }
    float dh1e[50];
    #pragma unroll
    for (int k=0;k<50;k++) dh1e[k]=0.f;
    float da[4] = {0.f,0.f,0.f,0.f};
    for (int g=0;g<GD_;g++){
      float acc=sB2[g];
      #pragma unroll
      for (int k=0;k<50;k++) acc += h1e[k]*sW2[k*GD_+g];
      const float Gg = tanhf(acc);
      float dG = 0.f;
      #pragma unroll
      for (int c=0;c<4;c++){
        const float dx = sDXA[c*GD_+g];
        dG    += sA[j*4+c]*dx;
        da[c] += Gg*dx;
      }
      const float dz = dG*inv*(1.f-Gg*Gg);
      #pragma unroll
      for (int k=0;k<50;k++) dh1e[k] += dz*sW2[k*GD_+g];
    }
    float dh0[25];
    #pragma unroll
    for (int i=0;i<25;i++) dh0[i]=0.f;
    #pragma unroll
    for (int k=0;k<50;k++){
      const float dz = dh1e[k]*(1.f-h1e[k]*h1e[k]);
      #pragma unroll
      for (int i=0;i<25;i++) dh0[i] += dz*sW1[i*50+k];
    }
    float dx0 = 0.f;
    #pragma unroll
    for (int i=0;i<25;i++){
      const float dz = dh0[i]*(1.f-h0[i]*h0[i]);
      dx0 += dz*sW0[0*25+i];
    }
    float dE[4];
    dE[0] = da[0]*inv + dx0;
    dE[1] = da[1]*inv;
    dE[2] = da[2]*inv;
    dE[3] = da[3]*inv;

    const size_t base = (size_t)n*J_ + j;
    float dEdx[3];
    #pragma unroll
    for (int d=0; d<3; d++){
      float acc=0.f;
      #pragma unroll
      for (int c=0;c<4;c++) acc += dE[c]*dfeat[(base*4+c)*3+d];
      dEdx[d]=acc;
      atomicAdd(&sF[d], -acc);
    }
    const int lnb = list_neigh[base];
    if (lnb > 0){
      int tgt = lnb - 1;
      tgt = tgt < 0 ? 0 : (tgt > N_-1 ? N_-1 : tgt);
      #pragma unroll
      for (int d=0; d<3; d++)
        atomicAdd(&out[1 + N_ + (size_t)tgt*3 + d], dEdx[d]);
      const float r0 = ImDR[base*3+0];
      const float r1 = ImDR[base*3+1];
      const float r2 = ImDR[base*3+2];
      atomicAdd(&sV[0], r0*dEdx[0]);
      atomicAdd(&sV[1], r0*dEdx[1]);
      atomicAdd(&sV[2], r0*dEdx[2]);
      atomicAdd(&sV[3], r1*dEdx[1]);
      atomicAdd(&sV[4], r1*dEdx[2]);
      atomicAdd(&sV[5], r2*dEdx[2]);
    }
  }
  __syncthreads();
  if (tid==0){
    #pragma unroll
    for (int d=0; d<3; d++)
      atomicAdd(&out[1 + N_ + (size_t)n*3 + d], sF[d]);
    float* V = out + 1 + N_ + 3*N_;
    atomicAdd(&V[0], sV[0]);
    atomicAdd(&V[1], sV[1]); atomicAdd(&V[3], sV[1]);
    atomicAdd(&V[2], sV[2]); atomicAdd(&V[6], sV[2]);
    atomicAdd(&V[4], sV[3]);
    atomicAdd(&V[5], sV[4]); atomicAdd(&V[7], sV[4]);
    atomicAdd(&V[8], sV[5]);
  }
}

// =====================================================================
extern "C" void kernel_launch(void* const* d_in, const int* in_sizes, int n_in,
                              void* d_out, int out_size, void* d_ws, size_t ws_size,
                              hipStream_t stream)
{
  (void)in_sizes; (void)n_in; (void)ws_size;
  const float* Ri    = (const float*)d_in[0];
  const float* dfeat = (const float*)d_in[1];
  const float* ImDR  = (const float*)d_in[2];
  const float* tv    = (const float*)d_in[3];
  const float* ew0   = (const float*)d_in[4];
  const float* eb0   = (const float*)d_in[5];
  const float* ew1   = (const float*)d_in[6];
  const float* eb1   = (const float*)d_in[7];
  const float* ew2   = (const float*)d_in[8];
  const float* eb2   = (const float*)d_in[9];
  const float* fw0   = (const float*)d_in[10];
  const float* fb0   = (const float*)d_in[11];
  const float* fw1   = (const float*)d_in[12];
  const float* fb1   = (const float*)d_in[13];
  const float* fw2   = (const float*)d_in[14];
  const float* fb2   = (const float*)d_in[15];
  const int*   lnb   = (const int*)d_in[16];
  float* out = (float*)d_out;

  float* ws  = (float*)d_ws;
  float* xa  = ws;                          // N*400
  float* dr  = xa  + (size_t)N_*4*GD_;      // N*1600
  float* h1  = dr  + (size_t)N_*D_;         // N*240
  float* h2  = h1  + (size_t)N_*FH_;        // N*240
  float* dz2 = h2  + (size_t)N_*FH_;        // N*240
  float* dz1 = dz2 + (size_t)N_*FH_;        // N*240
  float* ddr = dz1 + (size_t)N_*FH_;        // N*1600

  hipMemsetAsync(d_out, 0, (size_t)out_size*sizeof(float), stream);

  emb_fwd<<<N_, 256, 0, stream>>>(Ri, tv, ew0, eb0, ew1, eb1, ew2, eb2, xa, dr);
  fit_l0 <<<dim3(N_/16, 3), 32, 0, stream>>>(dr, fw0, fb0, h1);
  fit_l1 <<<N_/128, 256, 0, stream>>>(h1, fw1, fb1, h2);
  fit_out_k<<<N_, 32, 0, stream>>>(h2, fw2, fb2, out);
  dz2_k  <<<(N_*FH_+255)/256, 256, 0, stream>>>(h2, fw2, dz2);
  bwd_l1 <<<N_/128, 256, 0, stream>>>(dz2, fw1, h1, dz1);
  bwd_l0 <<<dim3(N_/16, 20), 32, 0, stream>>>(dz1, fw0, ddr);
  emb_bwd<<<N_, 256, 0, stream>>>(Ri, tv, ew0, eb0, ew1, eb1, ew2, eb2,
                                  dfeat, ImDR, lnb, xa, ddr, out);
}